// ReconDetectionConfidenceMap_42288247997085
// MI455X (gfx1250) — compile-verified
//
#include <hip/hip_runtime.h>
#include <math.h>

// V_WMMA_F32_16X16X4_F32 operand types (wave32):
//  A: 16x4 f32  -> 64 vals / 32 lanes = 2 VGPRs
//  B: 4x16 f32  -> 2 VGPRs
//  C/D: 16x16 f32 -> 8 VGPRs
typedef __attribute__((ext_vector_type(2))) float v2f;
typedef __attribute__((ext_vector_type(8))) float v8f;

#define BB 8
#define KK 64
#define HH 192
#define WW 256

__global__ __launch_bounds__(256)
void recon_conf_map_wmma(const float* __restrict__ kp, float* __restrict__ out)
{
    // Gaussian constants (compile-time)
    const float CORR = 0.3f, STD_X = 5.0f, STD_Y = 5.0f, EPS = 1e-5f;
    const float den  = 1.0f / (2.0f * 3.14159265358979323846f * STD_X * STD_Y *
                               sqrtf(1.0f - CORR * CORR));
    const float num1 = -1.0f / (2.0f * (1.0f - CORR * CORR));
    const float cy = 1.0f / (STD_Y * STD_Y);          // coeff of di^2
    const float cx = 1.0f / (STD_X * STD_X);          // coeff of dj^2
    const float cc = -2.0f * CORR / (STD_X * STD_Y);  // coeff of di*dj

    __shared__ float s_kp[KK * 2];

    const int tid  = threadIdx.x;
    const int wave = tid >> 5;
    const int lane = tid & 31;

    // one wave32 per 16x16 tile; 192 tiles per batch; 8 waves per block
    const int tiles_per_b = (HH / 16) * (WW / 16);     // 192 (divisible by 8)
    const int t   = blockIdx.x * 8 + wave;
    const int b   = t / tiles_per_b;                   // uniform per block
    const int rem = t % tiles_per_b;
    const int th  = rem / (WW / 16);
    const int tw  = rem % (WW / 16);

    // stage this batch's keypoints into LDS (512 B)
    if (tid < KK * 2) s_kp[tid] = kp[b * (KK * 2) + tid];
    __syncthreads();

    const int m    = lane & 15;   // A row M / B col N / D col N for this lane
    const int half = lane >> 4;   // 0: K={0,1} (A) / rows {0,1} (B) / M=r ; 1: K={2,3} / rows {2,3} / M=r+8

    const float yrow = (float)(th * 16 + m);  // image row for A's row m
    const float xcol = (float)(tw * 16 + m);  // image col for B's col n (= m)

    // ---- pass 1: rk = sum over k of score ----
    v8f rk = {};
    for (int k = 0; k < KK; ++k) {
        const float kx = s_kp[2 * k + 0];
        const float ky = s_kp[2 * k + 1];
        const float di = yrow - ky;
        const float dj = xcol - kx;

        // A[m,:] = [di^2, 1, di, 0] ; lanes 0-15 carry K=0,1 ; lanes 16-31 K=2,3
        v2f a;
        a.x = half ? di : di * di;
        a.y = half ? 0.0f : 1.0f;
        // B[:,n] = [cy, cx*dj^2, cc*dj, 0] ; lanes 0-15 rows 0,1 ; lanes 16-31 rows 2,3
        v2f bm;
        bm.x = half ? (cc * dj) : cy;
        bm.y = half ? 0.0f : (cx * dj * dj);

        v8f d = {};
        d = __builtin_amdgcn_wmma_f32_16x16x4_f32(
                /*neg_a=*/false, a, /*neg_b=*/false, bm,
                /*c_mod=*/(short)0, d, /*reuse_a=*/false, /*reuse_b=*/false);

#pragma unroll
        for (int r = 0; r < 8; ++r)
            rk[r] += den * __expf(num1 * d[r]);
    }
#pragma unroll
    for (int r = 0; r < 8; ++r)
        rk[r] = 1.0f / (rk[r] + EPS);

    // ---- pass 2: recompute score, normalize, store ----
    for (int k = 0; k < KK; ++k) {
        const float kx = s_kp[2 * k + 0];
        const float ky = s_kp[2 * k + 1];
        const float di = yrow - ky;
        const float dj = xcol - kx;

        v2f a;
        a.x = half ? di : di * di;
        a.y = half ? 0.0f : 1.0f;
        v2f bm;
        bm.x = half ? (cc * dj) : cy;
        bm.y = half ? 0.0f : (cx * dj * dj);

        v8f d = {};
        d = __builtin_amdgcn_wmma_f32_16x16x4_f32(
                false, a, false, bm, (short)0, d, false, false);

        // D element (M = r + 8*half, N = m); out[((b*K+k)*H + th*16+M)*W + tw*16+N]
        const size_t base = ((size_t)(b * KK + k) * HH + th * 16 + 8 * half) * WW
                            + tw * 16 + m;
#pragma unroll
        for (int r = 0; r < 8; ++r) {
            const float s = den * __expf(num1 * d[r]);
            out[base + (size_t)r * WW] = s * rk[r];
        }
    }
}

extern "C" void kernel_launch(void* const* d_in, const int* in_sizes, int n_in,
                              void* d_out, int out_size, void* d_ws, size_t ws_size,
                              hipStream_t stream) {
    // inputs: d_in[0]=batch_size (int scalar), d_in[1]=keypoints (f32 [8,64,2]),
    //         d_in[2]=DetectionMap (f32, shape-only -> never read)
    const float* kp = (const float*)d_in[1];
    float* out = (float*)d_out;
    (void)in_sizes; (void)n_in; (void)out_size; (void)d_ws; (void)ws_size;

    // 1536 tiles total, 8 waves/block -> 192 blocks of 256 threads
    recon_conf_map_wmma<<<192, 256, 0, stream>>>(kp, out);
}